// SageEncoder_69870527971698
// MI455X (gfx1250) — compile-verified
//
#include <hip/hip_runtime.h>
#include <math.h>

typedef __attribute__((ext_vector_type(2))) float v2f;
typedef __attribute__((ext_vector_type(8))) float v8f;

#define IN_F 128   // input feature dim (reference)
#define HID  512   // hidden dim (reference)

// ---------------------------------------------------------------------------
// 1) init workspace: mark[] = -1, accum/deg/count = 0
// ---------------------------------------------------------------------------
__global__ void init_ws(int* __restrict__ mark, int nNodes,
                        float* __restrict__ accum, int accumLen,
                        float* __restrict__ deg, int degLen,
                        int* __restrict__ count) {
  int i = blockIdx.x * blockDim.x + threadIdx.x;
  if (i < nNodes)  mark[i]  = -1;
  if (i < accumLen) accum[i] = 0.0f;
  if (i < degLen)  deg[i]   = 0.0f;
  if (i == 0)      *count   = 0;
}

// ---------------------------------------------------------------------------
// 2) assign compact slots to the unique nodes referenced by ids[]
// ---------------------------------------------------------------------------
__global__ void mark_ids(const int* __restrict__ ids, int B,
                         int* __restrict__ mark,
                         int* __restrict__ uniqueNodes,
                         int* __restrict__ count) {
  int i = blockIdx.x * blockDim.x + threadIdx.x;
  if (i >= B) return;
  int node = ids[i];
  int old = atomicCAS(&mark[node], -1, (int)0x80000000);
  if (old == -1) {
    int slot = atomicAdd(count, 1);
    uniqueNodes[slot] = node;
    atomicExch(&mark[node], slot);   // publish slot
  }
}

// ---------------------------------------------------------------------------
// 3) edge scatter restricted to marked destinations
//    accum[slot] += feat[src];  deg[slot] += 1
// ---------------------------------------------------------------------------
__global__ void edge_scatter(const int* __restrict__ src,
                             const int* __restrict__ dst,
                             int nE,
                             const float* __restrict__ feat,
                             const int* __restrict__ mark,
                             float* __restrict__ accum,
                             float* __restrict__ deg) {
  int e = blockIdx.x * blockDim.x + threadIdx.x;
  if (e >= nE) return;
  int s = mark[dst[e]];
  if (s < 0) return;                  // dst not in ids -> irrelevant
  int u = src[e];
  atomicAdd(&deg[s], 1.0f);
  const float4* fsrc = (const float4*)(feat + (size_t)u * IN_F);
  float* acc = accum + (size_t)s * IN_F;
#pragma unroll
  for (int j = 0; j < IN_F / 4; ++j) {
    float4 v = fsrc[j];
    atomicAdd(acc + 4 * j + 0, v.x);
    atomicAdd(acc + 4 * j + 1, v.y);
    atomicAdd(acc + 4 * j + 2, v.z);
    atomicAdd(acc + 4 * j + 3, v.w);
  }
}

// ---------------------------------------------------------------------------
// 4) h_neigh[slot] = (accum + feat[node]) / (deg + 1); zero for unused slots
// ---------------------------------------------------------------------------
__global__ void finalize_hneigh(const float* __restrict__ feat,
                                const int* __restrict__ uniqueNodes,
                                const int* __restrict__ count,
                                const float* __restrict__ accum,
                                const float* __restrict__ deg,
                                float* __restrict__ hneigh,
                                int Mpad) {
  int t = blockIdx.x * blockDim.x + threadIdx.x;
  int total = Mpad * (IN_F / 4);
  if (t >= total) return;
  int s = t / (IN_F / 4);
  int j = t % (IN_F / 4);
  int U = *count;
  float4 r = make_float4(0.f, 0.f, 0.f, 0.f);
  if (s < U) {
    int v = uniqueNodes[s];
    float inv = 1.0f / (deg[s] + 1.0f);
    float4 a = ((const float4*)(accum + (size_t)s * IN_F))[j];
    float4 f = ((const float4*)(feat + (size_t)v * IN_F))[j];
    r.x = (a.x + f.x) * inv;
    r.y = (a.y + f.y) * inv;
    r.z = (a.z + f.z) * inv;
    r.w = (a.w + f.w) * inv;
  }
  ((float4*)hneigh)[t] = r;
}

// ---------------------------------------------------------------------------
// 5) C[Mpad,512] = tanh(hneigh[Mpad,128] x W^T + bias), one 16x16 tile/wave,
//    V_WMMA_F32_16X16X4_F32 with K-loop of 32 steps.
//    A 16x4 frag: lane m=lane&15; K pair {k0,k0+1} (lanes 0-15) / {k0+2,k0+3}.
//    B 4x16 frag: lane n=lane&15; same K split. Bmat[k][n] = W[n*128 + k].
//    C/D: VGPR i -> row i (lanes 0-15) / row i+8 (lanes 16-31), col = lane&15.
// ---------------------------------------------------------------------------
__global__ void gemm_wmma_tanh(const float* __restrict__ hneigh,
                               const float* __restrict__ W,
                               const float* __restrict__ bias,
                               float* __restrict__ hout,
                               int Mtiles, int Ntiles) {
  int wave = threadIdx.x >> 5;
  int lane = threadIdx.x & 31;
  int tile = blockIdx.x * (blockDim.x >> 5) + wave;
  if (tile >= Mtiles * Ntiles) return;          // wave-uniform exit
  int tm = tile / Ntiles;
  int tn = tile % Ntiles;
  int half = lane >> 4;                          // 0 or 1
  int l15 = lane & 15;
  int mrow = tm * 16 + l15;
  int ncol = tn * 16 + l15;
  int kb = half * 2;                             // K offset within group of 4

  const float* arow = hneigh + (size_t)mrow * IN_F;
  const float* brow = W + (size_t)ncol * IN_F;   // row n of W == column n of B

  v8f c = {};
#pragma unroll
  for (int k0 = 0; k0 < IN_F; k0 += 4) {
    v2f a = *(const v2f*)(arow + k0 + kb);
    v2f b = *(const v2f*)(brow + k0 + kb);
    c = __builtin_amdgcn_wmma_f32_16x16x4_f32(
        /*neg_a=*/false, a, /*neg_b=*/false, b,
        /*c_mod=*/(short)0, c, /*reuse_a=*/false, /*reuse_b=*/false);
  }

  float bcol = bias[ncol];
  int rbase = tm * 16 + half * 8;
#pragma unroll
  for (int i = 0; i < 8; ++i) {
    hout[(size_t)(rbase + i) * HID + ncol] = tanhf(c[i] + bcol);
  }
}

// ---------------------------------------------------------------------------
// 6) out[i,:] = hout[mark[ids[i]], :]   (handles duplicate ids)
// ---------------------------------------------------------------------------
__global__ void gather_out(const int* __restrict__ ids,
                           const int* __restrict__ mark,
                           const float* __restrict__ hout,
                           float* __restrict__ out, int B) {
  int t = blockIdx.x * blockDim.x + threadIdx.x;
  int total = B * (HID / 4);
  if (t >= total) return;
  int i = t / (HID / 4);
  int j = t % (HID / 4);
  int s = mark[ids[i]];
  ((float4*)out)[t] = ((const float4*)(hout + (size_t)s * HID))[j];
}

// ---------------------------------------------------------------------------
extern "C" void kernel_launch(void* const* d_in, const int* in_sizes, int n_in,
                              void* d_out, int out_size, void* d_ws, size_t ws_size,
                              hipStream_t stream) {
  (void)n_in; (void)out_size; (void)ws_size;

  const float* feat  = (const float*)d_in[0];
  const float* W     = (const float*)d_in[1];
  const float* bias  = (const float*)d_in[2];
  const int*   src   = (const int*)d_in[3];
  const int*   dst   = (const int*)d_in[4];
  const int*   ids   = (const int*)d_in[5];

  const int nNodes = in_sizes[0] / IN_F;        // 50000
  const int nE     = in_sizes[3];               // 550000
  const int B      = in_sizes[5];               // 4096
  const int Mpad   = ((B + 15) / 16) * 16;      // slot rows, multiple of 16
  const int Mtiles = Mpad / 16;
  const int Ntiles = HID / 16;

  // workspace carve-out (256B aligned chunks)
  size_t off = 0;
  char* base = (char*)d_ws;
  auto take = [&](size_t bytes) -> void* {
    void* p = base + off;
    off += (bytes + 255) & ~(size_t)255;
    return p;
  };
  int*   mark   = (int*)  take((size_t)nNodes * 4);
  int*   uniq   = (int*)  take((size_t)B * 4);
  int*   count  = (int*)  take(16);
  float* deg    = (float*)take((size_t)Mpad * 4);
  float* accum  = (float*)take((size_t)Mpad * IN_F * 4);
  float* hneigh = (float*)take((size_t)Mpad * IN_F * 4);
  float* hout   = (float*)take((size_t)Mpad * HID * 4);

  float* out = (float*)d_out;

  const int T = 256;
  int accumLen = Mpad * IN_F;
  int initN = nNodes > accumLen ? nNodes : accumLen;

  init_ws<<<(initN + T - 1) / T, T, 0, stream>>>(mark, nNodes, accum, accumLen,
                                                 deg, Mpad, count);

  mark_ids<<<(B + T - 1) / T, T, 0, stream>>>(ids, B, mark, uniq, count);

  edge_scatter<<<(nE + T - 1) / T, T, 0, stream>>>(src, dst, nE, feat, mark,
                                                   accum, deg);

  int finTot = Mpad * (IN_F / 4);
  finalize_hneigh<<<(finTot + T - 1) / T, T, 0, stream>>>(feat, uniq, count,
                                                          accum, deg, hneigh, Mpad);

  int totalTiles = Mtiles * Ntiles;
  int wavesPerBlk = T / 32;
  gemm_wmma_tanh<<<(totalTiles + wavesPerBlk - 1) / wavesPerBlk, T, 0, stream>>>(
      hneigh, W, bias, hout, Mtiles, Ntiles);

  int gatTot = B * (HID / 4);
  gather_out<<<(gatTot + T - 1) / T, T, 0, stream>>>(ids, mark, hout, out, B);
}